// rnn_67697274519743
// MI455X (gfx1250) — compile-verified
//
#include <hip/hip_runtime.h>
#include <math.h>

// Problem dims (compile-time)
#define B_   128
#define T_   512
#define E_   128
#define H_   500
#define G4_  2000   // 4*H
#define D2_  1000   // 2*H

typedef __attribute__((ext_vector_type(2))) float v2f;
typedef __attribute__((ext_vector_type(8))) float v8f;

// CDNA5 fp32 WMMA: D(16x16,f32) = A(16x4,f32) * B(4x16,f32) + C
// 8-arg form: (neg_a, A, neg_b, B, c_mod, C, reuse_a, reuse_b)
__device__ __forceinline__ v8f wmma4(v2f a, v2f b, v8f c) {
  return __builtin_amdgcn_wmma_f32_16x16x4_f32(false, a, false, b, (short)0, c,
                                               false, false);
}

// ---------------------------------------------------------------------------
// Per-step gates GEMM: gates[dir][b][g] = x[b,t,:]@W_ih[g,:] + h[dir,b,:]@W_hh[g,:]
// One wave computes one 16(batch) x 16(gate) tile. 2000 tiles (2 dirs).
// Fragment layout (ISA 7.12.2, f32): lane = 16*half + q
//   A: a.x = A[q][2*half], a.y = A[q][2*half+1]      (contiguous float2 in K)
//   B: b.x = B[2*half][q], b.y = B[2*half+1][q] ; B[k][n] = W[n][k] -> contiguous float2
//   C/D: vgpr r = element (M = r + 8*half, N = q)
// ---------------------------------------------------------------------------
template <int I>
__global__ __launch_bounds__(256) void lstm_gates(
    const float* __restrict__ xin,    // [B, T, I]   (x for layer0, y0 for layer1)
    const float* __restrict__ wih_f,  // [4H, I]
    const float* __restrict__ whh_f,  // [4H, H]
    const float* __restrict__ wih_b,  // [4H, I]
    const float* __restrict__ whh_b,  // [4H, H]
    const float* __restrict__ hst,    // [2, B, H]  (this layer's h state)
    float* __restrict__ gates,        // [2, B, 4H]
    int s) {
  const int wave = threadIdx.x >> 5;
  const int lane = threadIdx.x & 31;
  const int q    = lane & 15;
  const int half = lane >> 4;
  const int kb   = half << 1;  // 0 or 2

  int tile = blockIdx.x * 8 + wave;  // 0..1999
  const int dir = (tile >= 1000) ? 1 : 0;
  tile -= dir * 1000;
  const int m0 = (tile / 125) * 16;  // batch tile base
  const int n0 = (tile % 125) * 16;  // gate tile base

  const int t = dir ? (T_ - 1 - s) : s;
  const float* wih = dir ? wih_b : wih_f;
  const float* whh = dir ? whh_b : whh_f;

  v8f acc = {};

  // Input contribution: K over I
  {
    const float* arow = xin + ((size_t)(m0 + q) * T_ + t) * I + kb;
    const float* brow = wih + (size_t)(n0 + q) * I + kb;
#pragma unroll 4
    for (int k = 0; k < I; k += 4) {
      v2f a = *(const v2f*)(arow + k);
      v2f b = *(const v2f*)(brow + k);
      acc = wmma4(a, b, acc);
    }
  }

  // Recurrent contribution: K over H
  {
    const float* arow = hst + ((size_t)dir * B_ + (m0 + q)) * H_ + kb;
    const float* brow = whh + (size_t)(n0 + q) * H_ + kb;
#pragma unroll 5
    for (int k = 0; k < H_; k += 4) {
      v2f a = *(const v2f*)(arow + k);
      v2f b = *(const v2f*)(brow + k);
      acc = wmma4(a, b, acc);
    }
  }

  // Store 16x16 tile
  float* gdst = gates + ((size_t)dir * B_ + m0 + half * 8) * G4_ + n0 + q;
#pragma unroll
  for (int r = 0; r < 8; ++r) gdst[(size_t)r * G4_] = acc[r];
}

// ---------------------------------------------------------------------------
// Per-step cell update: activations, masked h/c update, y write.
// One thread per (dir, b, h): 2*128*500 = 128000 threads.
// ---------------------------------------------------------------------------
__global__ __launch_bounds__(256) void lstm_cell(
    const float* __restrict__ gates,  // [2, B, 4H]
    const float* __restrict__ bih_f, const float* __restrict__ bhh_f,
    const float* __restrict__ bih_b, const float* __restrict__ bhh_b,
    float* __restrict__ hst, float* __restrict__ cst,  // [2, B, H]
    const int* __restrict__ lens,                       // [B]
    float* __restrict__ y,                              // [B, T, 2H]
    int s) {
  int id = blockIdx.x * 256 + threadIdx.x;  // < 2*B*H
  const int dir = id / (B_ * H_);
  int r = id - dir * (B_ * H_);
  const int b = r / H_;
  const int h = r - b * H_;

  const int t = dir ? (T_ - 1 - s) : s;
  const float m = (t < lens[b]) ? 1.0f : 0.0f;

  const float* bih = dir ? bih_b : bih_f;
  const float* bhh = dir ? bhh_b : bhh_f;

  const size_t gb = ((size_t)dir * B_ + b) * G4_ + h;
  float gi = gates[gb]          + bih[h]          + bhh[h];
  float gf = gates[gb + H_]     + bih[h + H_]     + bhh[h + H_];
  float gg = gates[gb + 2 * H_] + bih[h + 2 * H_] + bhh[h + 2 * H_];
  float go = gates[gb + 3 * H_] + bih[h + 3 * H_] + bhh[h + 3 * H_];

  float i = 1.0f / (1.0f + __expf(-gi));
  float f = 1.0f / (1.0f + __expf(-gf));
  float g = tanhf(gg);
  float o = 1.0f / (1.0f + __expf(-go));

  const size_t st = (size_t)dir * (B_ * H_) + (size_t)b * H_ + h;
  float c_old = cst[st];
  float h_old = hst[st];
  float c_new = f * c_old + i * g;
  float h_new = o * tanhf(c_new);
  hst[st] = m * h_new + (1.0f - m) * h_old;
  cst[st] = m * c_new + (1.0f - m) * c_old;
  y[((size_t)b * T_ + t) * D2_ + (size_t)dir * H_ + h] = m * h_new;
}

__global__ __launch_bounds__(256) void zero_f32(float* p, int n) {
  int i = blockIdx.x * 256 + threadIdx.x;
  if (i < n) p[i] = 0.0f;
}

__global__ __launch_bounds__(256) void copy_f32(const float* __restrict__ s,
                                                float* __restrict__ d, int n) {
  int i = blockIdx.x * 256 + threadIdx.x;
  if (i < n) d[i] = s[i];
}

// ---------------------------------------------------------------------------
extern "C" void kernel_launch(void* const* d_in, const int* in_sizes, int n_in,
                              void* d_out, int out_size, void* d_ws,
                              size_t ws_size, hipStream_t stream) {
  (void)in_sizes; (void)n_in; (void)out_size; (void)ws_size;

  const float* x   = (const float*)d_in[0];
  const int* lens  = (const int*)d_in[1];
  const float* W[16];
  for (int i = 0; i < 16; ++i) W[i] = (const float*)d_in[2 + i];
  // W layout per (layer, dir): {w_ih, w_hh, b_ih, b_hh}
  //   idx base = 8*layer + 4*dir

  float* out   = (float*)d_out;
  float* y_out = out;                                  // [B,T,2H]
  float* h_out = out + (size_t)B_ * T_ * D2_;          // [4,B,H] then c [4,B,H]

  float* ws     = (float*)d_ws;
  float* y0     = ws;                                  // [B,T,2H] layer-0 output
  float* hstate = y0 + (size_t)B_ * T_ * D2_;          // [2 layers][2 dirs][B][H]
  float* cstate = hstate + 4 * B_ * H_;                // same shape, contiguous
  float* gates  = cstate + 4 * B_ * H_;                // [2][B][4H]

  // zero h and c states (contiguous: 2 * 4*B*H floats)
  {
    int n = 2 * 4 * B_ * H_;
    zero_f32<<<(n + 255) / 256, 256, 0, stream>>>(hstate, n);
  }

  for (int layer = 0; layer < 2; ++layer) {
    const int base = 8 * layer;
    const float* wih_f = W[base + 0];
    const float* whh_f = W[base + 1];
    const float* bih_f = W[base + 2];
    const float* bhh_f = W[base + 3];
    const float* wih_b = W[base + 4];
    const float* whh_b = W[base + 5];
    const float* bih_b = W[base + 6];
    const float* bhh_b = W[base + 7];

    float* hL = hstate + (size_t)layer * 2 * B_ * H_;
    float* cL = cstate + (size_t)layer * 2 * B_ * H_;
    const float* xin = layer ? y0 : x;
    float* yDst = layer ? y_out : y0;

    for (int s = 0; s < T_; ++s) {
      if (layer == 0) {
        lstm_gates<E_><<<250, 256, 0, stream>>>(xin, wih_f, whh_f, wih_b,
                                                whh_b, hL, gates, s);
      } else {
        lstm_gates<D2_><<<250, 256, 0, stream>>>(xin, wih_f, whh_f, wih_b,
                                                 whh_b, hL, gates, s);
      }
      lstm_cell<<<500, 256, 0, stream>>>(gates, bih_f, bhh_f, bih_b, bhh_b,
                                         hL, cL, lens, yDst, s);
    }
  }

  // Final h,c: state layout [2*layer+dir][B][H] == reference stack order
  // [l0f, l0b, l1f, l1b]; h and c regions are contiguous in both src and dst.
  {
    int n = 2 * 4 * B_ * H_;
    copy_f32<<<(n + 255) / 256, 256, 0, stream>>>(hstate, h_out, n);
  }
}